// PairNet_76501957476437
// MI455X (gfx1250) — compile-verified
//
#include <hip/hip_runtime.h>

typedef __attribute__((ext_vector_type(2))) float v2f;
typedef __attribute__((ext_vector_type(4))) float v4f;
typedef __attribute__((ext_vector_type(8))) float v8f;

#define D_MODEL 128
#define D_EMB   64
#define L_SEQ   768
#define BATCH   2
#define MAXLEN  32
#define ROWS    (BATCH * L_SEQ)   // 1536

// ---------------------------------------------------------------------------
// Kernel 1: tiny fp32 GEMMs via V_WMMA_F32_16X16X4_F32.
//   P[r,c] = sum_k E[r,k] * W_proj[k,   c] + (b_proj[c] + b_pos[c])
//   Q[r,c] = sum_k E[r,k] * W_proj[64+k,c]
//   E[r,k] = (tok!=0) ? emb[tok*64+k] : 0,  tok = msa_tokens[(r/768)*3072 + r%768]
// One wave (32 lanes) per 16-row tile. EXEC is all-1s (blockDim == 32).
// A 16x4 f32 layout: lane%16 = M; lanes<16 hold K={0,1}, lanes>=16 hold K={2,3}.
// B 4x16 f32 layout (mirror): lane%16 = N; VGPR pair = K pair by lane/16.
// D 16x16 f32 layout: VGPR v -> M = v + 8*(lane/16), N = lane%16.
// ---------------------------------------------------------------------------
__global__ __launch_bounds__(32)
void pairnet_proj_wmma(const int*   __restrict__ msa_tokens,
                       const float* __restrict__ emb,
                       const float* __restrict__ W_proj,
                       const float* __restrict__ b_proj,
                       const float* __restrict__ b_pos,
                       float* __restrict__ P,
                       float* __restrict__ Q) {
    const int lane   = threadIdx.x;       // 0..31
    const int m0     = blockIdx.x * 16;   // row tile base
    const int rlocal = lane & 15;
    const int half   = lane >> 4;         // 0 or 1
    const int kb     = half * 2;          // K sub-offset: 0 or 2

    // Gather this lane's A-matrix row source.
    const int   r    = m0 + rlocal;
    const int   bi   = r / L_SEQ;
    const int   li   = r - bi * L_SEQ;
    const int   tok  = msa_tokens[bi * (4 * L_SEQ) + li];   // msa_tokens[b,0,l]
    const float mask = (tok != 0) ? 1.0f : 0.0f;
    const float* erow = emb + tok * D_EMB;

    // Preload all 16 A-fragments (K = 0..63 in steps of 4); reused for 8 N-tiles x {P,Q}.
    v2f afrag[16];
#pragma unroll
    for (int ks = 0; ks < 16; ++ks) {
        const int k = ks * 4 + kb;
        afrag[ks].x = erow[k]     * mask;
        afrag[ks].y = erow[k + 1] * mask;
    }

#pragma unroll
    for (int nt = 0; nt < 8; ++nt) {
        const int col = nt * 16 + rlocal;   // N index for this lane
        v8f accP = {};
        v8f accQ = {};
#pragma unroll
        for (int ks = 0; ks < 16; ++ks) {
            const int k = ks * 4 + kb;
            v2f bP, bQ;
            bP.x = W_proj[(k     ) * D_MODEL + col];
            bP.y = W_proj[(k +  1) * D_MODEL + col];
            bQ.x = W_proj[(k + 64) * D_MODEL + col];
            bQ.y = W_proj[(k + 65) * D_MODEL + col];
            // 8 args: (neg_a, A, neg_b, B, c_mod, C, reuse_a, reuse_b)
            accP = __builtin_amdgcn_wmma_f32_16x16x4_f32(false, afrag[ks], false, bP,
                                                         (short)0, accP, false, false);
            accQ = __builtin_amdgcn_wmma_f32_16x16x4_f32(false, afrag[ks], false, bQ,
                                                         (short)0, accQ, false, false);
        }
        const float bias = b_proj[col] + b_pos[col];  // fold both biases into P
#pragma unroll
        for (int v = 0; v < 8; ++v) {
            const int row = m0 + v + 8 * half;        // D-matrix M mapping
            P[row * D_MODEL + col] = accP[v] + bias;
            Q[row * D_MODEL + col] = accQ[v];
        }
    }
}

// ---------------------------------------------------------------------------
// Kernel 2: streaming broadcast-add (write-BW bound; ~604 MB @ 23.3 TB/s).
//   out[b,i,j,c] = P[b,i,c] + Q[b,j,c] + W_pos[clip(i-j,±31)+31, c]
// Block = 256 threads = 8 j-values x 32 lanes; each lane does one float4
// (16 B) -> 512 B contiguous per 32-lane group. Non-temporal stores keep the
// 604 MB stream out of L2 so P/Q/W_pos stay resident.
// ---------------------------------------------------------------------------
__global__ __launch_bounds__(256)
void pairnet_assemble(const float* __restrict__ P,
                      const float* __restrict__ Q,
                      const float* __restrict__ W_pos,
                      float* __restrict__ out) {
    const int tid = threadIdx.x;
    const int jo  = tid >> 5;           // 0..7
    const int c4  = (tid & 31) * 4;     // channel base (float4)

    int bid = blockIdx.x;
    const int j = (bid % (L_SEQ / 8)) * 8 + jo;
    bid /= (L_SEQ / 8);
    const int i = bid % L_SEQ;
    const int b = bid / L_SEQ;

    int rel = i - j;
    rel = (rel < -(MAXLEN - 1)) ? -(MAXLEN - 1) : rel;
    rel = (rel >  (MAXLEN - 1)) ?  (MAXLEN - 1) : rel;
    rel += (MAXLEN - 1);

    const v4f p = *(const v4f*)(P     + (size_t)(b * L_SEQ + i) * D_MODEL + c4);
    const v4f q = *(const v4f*)(Q     + (size_t)(b * L_SEQ + j) * D_MODEL + c4);
    const v4f w = *(const v4f*)(W_pos + (size_t)rel * D_MODEL + c4);
    const v4f o = p + q + w;

    const size_t off = (((size_t)(b * L_SEQ + i) * L_SEQ) + j) * D_MODEL + c4;
    __builtin_nontemporal_store(o, (v4f*)(out + off));
}

extern "C" void kernel_launch(void* const* d_in, const int* in_sizes, int n_in,
                              void* d_out, int out_size, void* d_ws, size_t ws_size,
                              hipStream_t stream) {
    const int*   msa    = (const int*)  d_in[0];  // (2,4,768)
    const float* emb    = (const float*)d_in[1];  // (5,64)
    const float* W_proj = (const float*)d_in[2];  // (128,128)
    const float* b_proj = (const float*)d_in[3];  // (128,)
    const float* W_pos  = (const float*)d_in[4];  // (63,128)
    const float* b_pos  = (const float*)d_in[5];  // (128,)
    float* out = (float*)d_out;                   // (2,768,768,128)

    float* P = (float*)d_ws;                      // 1536*128 floats
    float* Q = P + (size_t)ROWS * D_MODEL;        // 1536*128 floats (total ~1.5 MB)

    pairnet_proj_wmma<<<ROWS / 16, 32, 0, stream>>>(msa, emb, W_proj, b_proj, b_pos, P, Q);

    const int nblocks = BATCH * L_SEQ * (L_SEQ / 8);  // 147,456
    pairnet_assemble<<<nblocks, 256, 0, stream>>>(P, Q, W_pos, out);
}